// AttentionModule_15728170238431
// MI455X (gfx1250) — compile-verified
//
#include <hip/hip_runtime.h>
#include <hip/hip_bf16.h>
#include <math.h>

#define Dm 64
typedef __attribute__((ext_vector_type(2))) float v2f;
typedef __attribute__((ext_vector_type(8))) float v8f;

#if defined(__has_builtin)
#if __has_builtin(__builtin_amdgcn_tanhf)
#define FAST_TANH(x) __builtin_amdgcn_tanhf(x)
#endif
#if __has_builtin(__builtin_amdgcn_rcpf)
#define FAST_RCP(x) __builtin_amdgcn_rcpf(x)
#endif
#endif
#ifndef FAST_TANH
#define FAST_TANH(x) tanhf(x)
#endif
#ifndef FAST_RCP
#define FAST_RCP(x) (1.0f / (x))
#endif

// ---------------------------------------------------------------- zeroing
__global__ void zero_kernel(float* __restrict__ p, int n) {
    int i = blockIdx.x * blockDim.x + threadIdx.x;
    if (i < n) p[i] = 0.0f;
}

// ---------------------------------------------------------------- pass 1: segment sum + counts
// One wave per contiguous row chunk; lane owns cols {2l, 2l+1} as float2
// (coalesced 256B global_load_b64 per wave step). batch ids are loaded 32 at
// a time (one coalesced load per chunk) and broadcast via shfl. Segment
// accumulated in registers; atomics only when the (sorted) batch id changes.
__global__ void scatter_sum_kernel(const float* __restrict__ x,
                                   const int* __restrict__ batch,
                                   float* __restrict__ seg_sum,
                                   float* __restrict__ counts,
                                   int N, int rows_per_wave) {
    const int lane = threadIdx.x & 31;
    const int wave = blockIdx.x * (blockDim.x >> 5) + (threadIdx.x >> 5);
    const long long row0 = (long long)wave * rows_per_wave;
    if (row0 >= N) return;

    const v2f* __restrict__ x2 = (const v2f*)x;

    int cur = -1;
    int cnt = 0;
    v2f acc = {0.0f, 0.0f};

    for (int rb = 0; rb < rows_per_wave; rb += 32) {
        const long long rowb = row0 + rb;
        if (rowb >= N) break;
        const long long rem = N - rowb;
        const int lim = rem < 32 ? (int)rem : 32;

        long long bi = rowb + lane;
        int bv = batch[bi < N ? bi : (N - 1)];

        if (rowb + 32 < N)
            __builtin_prefetch(&x[(rowb + 32) * Dm + 2 * lane], 0, 1);

        for (int r = 0; r < lim; ++r) {
            int g = __shfl(bv, r, 32);
            if (g != cur) {
                if (cur >= 0) {
                    atomicAdd(&seg_sum[(long long)cur * Dm + 2 * lane],     acc.x);
                    atomicAdd(&seg_sum[(long long)cur * Dm + 2 * lane + 1], acc.y);
                    if (lane == 0) atomicAdd(&counts[cur], (float)cnt);
                }
                cur = g; cnt = 0;
                acc.x = 0.0f; acc.y = 0.0f;
            }
            v2f xv = x2[(rowb + r) * 32 + lane];
            acc.x += xv.x;
            acc.y += xv.y;
            ++cnt;
        }
    }
    if (cur >= 0) {
        atomicAdd(&seg_sum[(long long)cur * Dm + 2 * lane],     acc.x);
        atomicAdd(&seg_sum[(long long)cur * Dm + 2 * lane + 1], acc.y);
        if (lane == 0) atomicAdd(&counts[cur], (float)cnt);
    }
}

// ---------------------------------------------------------------- pass 2: tanh(mean @ W) via WMMA f32 16x16x4
// One wave per 16-row tile of mean. A(16x4) frag: lanes 0-15 -> K={k0,k0+1},
// lanes 16-31 -> K={k0+2,k0+3}; B(4x16) mirrored; C/D(16x16): vgpr v = row
// v (+8 for upper lanes), lane%16 = col.
__global__ void mean_matmul_tanh_kernel(const float* __restrict__ seg_sum,
                                        const float* __restrict__ counts,
                                        const float* __restrict__ W,
                                        float* __restrict__ tg,
                                        int G) {
    const int lane  = threadIdx.x & 31;
    const int wave  = blockIdx.x * (blockDim.x >> 5) + (threadIdx.x >> 5);
    const int m0    = wave * 16;
    if (m0 >= G) return;

    const int mrow  = m0 + (lane & 15);
    const int khalf = lane >> 4;                    // 0 or 1
    const float inv = FAST_RCP(fmaxf(counts[mrow], 1.0f));

    // Preload the A fragments for all 16 K-chunks (this lane's 32 floats of the row).
    v2f afrag[16];
#pragma unroll
    for (int kk = 0; kk < 16; ++kk) {
        int k = kk * 4 + 2 * khalf;
        afrag[kk].x = seg_sum[(long long)mrow * Dm + k]     * inv;
        afrag[kk].y = seg_sum[(long long)mrow * Dm + k + 1] * inv;
    }

    const int colN = lane & 15;
#pragma unroll
    for (int n = 0; n < 4; ++n) {
        v8f c = {};
        const int colB = n * 16 + colN;
#pragma unroll
        for (int kk = 0; kk < 16; ++kk) {
            int k = kk * 4 + 2 * khalf;
            v2f b;
            b.x = W[k * Dm + colB];
            b.y = W[(k + 1) * Dm + colB];
            c = __builtin_amdgcn_wmma_f32_16x16x4_f32(
                    /*neg_a=*/false, afrag[kk], /*neg_b=*/false, b,
                    /*c_mod=*/(short)0, c, /*reuse_a=*/false, /*reuse_b=*/false);
        }
        const int rbase = m0 + 8 * khalf;
#pragma unroll
        for (int v = 0; v < 8; ++v)
            tg[(long long)(rbase + v) * Dm + n * 16 + colN] = FAST_TANH(c[v]);
    }
}

// ---------------------------------------------------------------- pass 3: coefs + weighted scatter-add
// One wave per row chunk; lane owns cols {2l, 2l+1} as a float2. Dot product
// via wave32 shfl_xor tree; sigmoid via v_exp + v_rcp; tg row cached in
// registers per segment; weighted accumulation in registers, atomics only at
// segment boundaries. batch ids chunk-loaded + shfl-broadcast.
__global__ void attn_scatter_kernel(const float* __restrict__ x,
                                    const int* __restrict__ batch,
                                    const float* __restrict__ tg,
                                    float* __restrict__ out,
                                    int N, int rows_per_wave) {
    const int lane = threadIdx.x & 31;
    const int wave = blockIdx.x * (blockDim.x >> 5) + (threadIdx.x >> 5);
    const long long row0 = (long long)wave * rows_per_wave;
    if (row0 >= N) return;

    const v2f* __restrict__ x2  = (const v2f*)x;
    const v2f* __restrict__ tg2 = (const v2f*)tg;

    int cur = -1;
    v2f tw  = {0.0f, 0.0f};
    v2f acc = {0.0f, 0.0f};

    for (int rb = 0; rb < rows_per_wave; rb += 32) {
        const long long rowb = row0 + rb;
        if (rowb >= N) break;
        const long long rem = N - rowb;
        const int lim = rem < 32 ? (int)rem : 32;

        long long bi = rowb + lane;
        int bv = batch[bi < N ? bi : (N - 1)];

        if (rowb + 32 < N)
            __builtin_prefetch(&x[(rowb + 32) * Dm + 2 * lane], 0, 1);

        for (int r = 0; r < lim; ++r) {
            int g = __shfl(bv, r, 32);
            if (g != cur) {
                if (cur >= 0) {
                    atomicAdd(&out[(long long)cur * Dm + 2 * lane],     acc.x);
                    atomicAdd(&out[(long long)cur * Dm + 2 * lane + 1], acc.y);
                }
                cur = g;
                acc.x = 0.0f; acc.y = 0.0f;
                tw = tg2[(long long)g * 32 + lane];
            }

            v2f xv = x2[(rowb + r) * 32 + lane];

            float p = xv.x * tw.x + xv.y * tw.y;
#pragma unroll
            for (int off = 16; off > 0; off >>= 1)
                p += __shfl_xor(p, off, 32);

            float coef = FAST_RCP(1.0f + __expf(-p));
            acc.x += coef * xv.x;
            acc.y += coef * xv.y;
        }
    }
    if (cur >= 0) {
        atomicAdd(&out[(long long)cur * Dm + 2 * lane],     acc.x);
        atomicAdd(&out[(long long)cur * Dm + 2 * lane + 1], acc.y);
    }
}

// ---------------------------------------------------------------- launcher
extern "C" void kernel_launch(void* const* d_in, const int* in_sizes, int n_in,
                              void* d_out, int out_size, void* d_ws, size_t ws_size,
                              hipStream_t stream) {
    const float* x     = (const float*)d_in[0];
    const int*   batch = (const int*)d_in[1];
    // d_in[2] is the scalar `size`; G is recovered from out_size instead.
    const float* W     = (const float*)d_in[3];

    const int N = in_sizes[0] / Dm;       // 2,000,000
    const int G = out_size   / Dm;        // 8,192

    float* seg_sum = (float*)d_ws;                      // G*Dm floats
    float* counts  = seg_sum + (size_t)G * Dm;          // G floats (contiguous)
    float* tg      = counts + G;                        // G*Dm floats
    float* out     = (float*)d_out;

    // Zero accumulators (seg_sum + counts contiguous) + output.
    {
        int nacc = G * Dm + G;
        int nout = G * Dm;
        zero_kernel<<<(nacc + 255) / 256, 256, 0, stream>>>(seg_sum, nacc);
        zero_kernel<<<(nout + 255) / 256, 256, 0, stream>>>(out, nout);
    }

    // Pass 1: segment sums + counts (wave-per-chunk, float2 lanes).
    {
        const int rows_per_wave = 256;
        int waves  = (N + rows_per_wave - 1) / rows_per_wave;
        int blocks = (waves + 7) / 8;     // 8 waves per 256-thread block
        scatter_sum_kernel<<<blocks, 256, 0, stream>>>(x, batch, seg_sum, counts,
                                                       N, rows_per_wave);
    }

    // Pass 2: tg = tanh(mean @ W) — WMMA f32 16x16x4, one wave per 16 rows.
    {
        int waves  = G / 16;              // 512
        int blocks = waves / 8;           // 256 threads = 8 waves per block
        mean_matmul_tanh_kernel<<<blocks, 256, 0, stream>>>(seg_sum, counts, W, tg, G);
    }

    // Pass 3: attention coefficients + weighted scatter-add.
    {
        const int rows_per_wave = 256;
        int waves  = (N + rows_per_wave - 1) / rows_per_wave;
        int blocks = (waves + 7) / 8;     // 8 waves per 256-thread block
        attn_scatter_kernel<<<blocks, 256, 0, stream>>>(x, batch, tg, out,
                                                        N, rows_per_wave);
    }
}